// Encoder_PPI_62663572848808
// MI455X (gfx1250) — compile-verified
//
#include <hip/hip_runtime.h>

typedef __attribute__((ext_vector_type(2))) float v2f;
typedef __attribute__((ext_vector_type(8))) float v8f;

#define N_CH 128

// ---------------------------------------------------------------------------
// 1) zero the output (atomic accumulation target, float4-vectorized) and
//    init deg[i]=1 (self-loop contribution)
// ---------------------------------------------------------------------------
__global__ void zero_init_kernel(float4* __restrict__ out4, float* __restrict__ deg,
                                 int n_nodes) {
    long long i = (long long)blockIdx.x * blockDim.x + threadIdx.x;
    long long total4 = (long long)n_nodes * (N_CH / 4);
    if (i < total4) out4[i] = make_float4(0.f, 0.f, 0.f, 0.f);
    if (i < n_nodes) deg[i] = 1.0f;
}

// ---------------------------------------------------------------------------
// 2) degree: hardware fp32 atomic add per edge destination
// ---------------------------------------------------------------------------
__global__ void degree_kernel(const long long* __restrict__ ei,
                              float* __restrict__ deg, int n_edges) {
    int e = blockIdx.x * blockDim.x + threadIdx.x;
    if (e >= n_edges) return;
    long long d = ei[(long long)n_edges + e];  // edge_index[1][e]
    unsafeAtomicAdd(&deg[d], 1.0f);
}

// ---------------------------------------------------------------------------
// 3) deg -> d^-1/2 in place (deg >= 1 always due to self-loops)
// ---------------------------------------------------------------------------
__global__ void rsqrt_kernel(float* __restrict__ deg, int n_nodes) {
    int i = blockIdx.x * blockDim.x + threadIdx.x;
    if (i < n_nodes) deg[i] = rsqrtf(deg[i]);
}

// ---------------------------------------------------------------------------
// 4) xw = x @ W via V_WMMA_F32_16X16X4_F32 (exact fp32).
//    Block = 256 threads = 8 waves; block covers 128 rows; wave w owns rows
//    [base+16w, base+16w+15] x all 128 columns (8 N-tiles, 64 acc VGPRs).
//
//    W is staged in LDS as K-pairs with an XOR-32 bank swizzle:
//      element (k, n) -> dword  kp*256 + ((2n) ^ ((kp&1)<<5)),  kp = k>>1
//    so each B operand {W[ka][n], W[ka+1][n]} is ONE aligned ds_load_b64,
//    and the two half-wave lane groups (rows kp / kp+1) hit disjoint
//    32-bank halves -> conflict-free. Exactly 64 KB of the 320 KB/WGP LDS.
//
//    A rows are clamped (not masked): garbage rows only affect WMMA output
//    rows that the guarded store discards, so the K-loop has no branches.
// ---------------------------------------------------------------------------
__global__ __launch_bounds__(256) void gemm_wmma_kernel(
    const float* __restrict__ x, const float* __restrict__ W,
    float* __restrict__ xw, int n_nodes) {
    __shared__ float Wl[64 * 256];  // 64 KB

    const int lane = threadIdx.x & 31;
    const int wave = threadIdx.x >> 5;
    const int m  = lane & 15;   // row (A) / col (B,D) within 16x16 tile
    const int kq = lane >> 4;   // which K pair of the 4-wide K slab
    const int row_base = blockIdx.x * 128 + wave * 16;
    const int row = row_base + m;
    const int crow = row < n_nodes ? row : (n_nodes - 1);
    const float* __restrict__ xrow = x + (size_t)crow * N_CH;

    // get this wave's A row moving while we stage W (global_prefetch_b8)
    __builtin_prefetch(xrow, 0, 0);
    __builtin_prefetch(xrow + 64, 0, 0);

    // stage W (128x128) into LDS, K-paired + swizzled
    for (int idx = threadIdx.x; idx < 64 * 128; idx += 256) {
        int n  = idx & 127;
        int kp = idx >> 7;  // 0..63
        float2 w2;
        w2.x = W[(2 * kp) * N_CH + n];
        w2.y = W[(2 * kp + 1) * N_CH + n];
        int off = kp * 256 + ((2 * n) ^ ((kp & 1) << 5));
        *(float2*)&Wl[off] = w2;
    }
    __syncthreads();

    v8f acc[8];
#pragma unroll
    for (int t = 0; t < 8; ++t) acc[t] = (v8f)0.0f;

    for (int k0 = 0; k0 < N_CH; k0 += 4) {
        float2 a2 = *(const float2*)(xrow + k0 + kq * 2);
        v2f A;
        A.x = a2.x;
        A.y = a2.y;
        const int kp = (k0 >> 1) + kq;               // even + kq -> kp&1 == kq
        const float* Bbase = &Wl[kp * 256 + 2 * m];  // + ((t^kq)*32) per tile
#pragma unroll
        for (int t = 0; t < 8; ++t) {
            float2 b2 = *(const float2*)(Bbase + ((t ^ kq) << 5));
            v2f B;
            B.x = b2.x;
            B.y = b2.y;
            acc[t] = __builtin_amdgcn_wmma_f32_16x16x4_f32(
                false, A, false, B, (short)0, acc[t], false, false);
        }
    }

#pragma unroll
    for (int t = 0; t < 8; ++t) {
#pragma unroll
        for (int i = 0; i < 8; ++i) {
            int r = row_base + kq * 8 + i;
            if (r < n_nodes) xw[(size_t)r * N_CH + t * 16 + m] = acc[t][i];
        }
    }
}

// ---------------------------------------------------------------------------
// 5) edge scatter: one wave per edge; lane handles 4 channels (coalesced
//    512B float4 gather of xw[src], 4 hardware fp32 atomic adds to out[dst]).
//    xw (51MB) and out (51MB) both fit in the 192MB L2 -> L2-rate bound.
// ---------------------------------------------------------------------------
__global__ __launch_bounds__(256) void scatter_kernel(
    const long long* __restrict__ ei, const float* __restrict__ xw,
    const float* __restrict__ dinv, float* __restrict__ out, int n_edges) {
    const int lane = threadIdx.x & 31;
    const long long gwave =
        (((long long)blockIdx.x * blockDim.x) + threadIdx.x) >> 5;
    if (gwave >= n_edges) return;
    const long long s = ei[gwave];                       // src
    const long long d = ei[(long long)n_edges + gwave];  // dst
    const float norm = dinv[s] * dinv[d];
    const float4 v = *(const float4*)(xw + s * N_CH + lane * 4);
    float* o = out + d * N_CH + lane * 4;
    unsafeAtomicAdd(o + 0, v.x * norm);
    unsafeAtomicAdd(o + 1, v.y * norm);
    unsafeAtomicAdd(o + 2, v.z * norm);
    unsafeAtomicAdd(o + 3, v.w * norm);
}

// ---------------------------------------------------------------------------
// 6) finalize: add self-loop term xw * dinv^2, bias, per-channel PReLU
// ---------------------------------------------------------------------------
__global__ void finalize_kernel(float* __restrict__ out,
                                const float* __restrict__ xw,
                                const float* __restrict__ dinv,
                                const float* __restrict__ b,
                                const float* __restrict__ a, int n_nodes) {
    long long i = (long long)blockIdx.x * blockDim.x + threadIdx.x;
    if (i >= (long long)n_nodes * N_CH) return;
    int node = (int)(i >> 7);
    int c = (int)(i & (N_CH - 1));
    float di = dinv[node];
    float v = out[i] + xw[i] * (di * di) + b[c];
    out[i] = (v >= 0.0f) ? v : a[c] * v;
}

// ---------------------------------------------------------------------------
extern "C" void kernel_launch(void* const* d_in, const int* in_sizes, int n_in,
                              void* d_out, int out_size, void* d_ws, size_t ws_size,
                              hipStream_t stream) {
    const float*     x  = (const float*)d_in[0];
    const long long* ei = (const long long*)d_in[1];  // int64 (2, E) row-major
    const float*     W  = (const float*)d_in[2];
    const float*     b  = (const float*)d_in[3];
    const float*     a  = (const float*)d_in[4];
    float*           out = (float*)d_out;

    const int n_nodes = in_sizes[0] / N_CH;
    const int n_edges = in_sizes[1] / 2;

    // workspace: xw [n_nodes*128] floats, then deg/dinv [n_nodes] floats
    float* xw  = (float*)d_ws;
    float* deg = xw + (size_t)n_nodes * N_CH;

    const int tb = 256;
    const long long total4 = (long long)n_nodes * (N_CH / 4);

    zero_init_kernel<<<(unsigned)((total4 + tb - 1) / tb), tb, 0, stream>>>(
        (float4*)out, deg, n_nodes);
    degree_kernel<<<(n_edges + tb - 1) / tb, tb, 0, stream>>>(ei, deg, n_edges);
    rsqrt_kernel<<<(n_nodes + tb - 1) / tb, tb, 0, stream>>>(deg, n_nodes);
    gemm_wmma_kernel<<<(n_nodes + 127) / 128, 256, 0, stream>>>(x, W, xw, n_nodes);
    const long long sc_threads = (long long)n_edges * 32;
    scatter_kernel<<<(unsigned)((sc_threads + tb - 1) / tb), tb, 0, stream>>>(
        ei, xw, deg, out, n_edges);
    const long long total = (long long)n_nodes * N_CH;
    finalize_kernel<<<(unsigned)((total + tb - 1) / tb), tb, 0, stream>>>(
        out, xw, deg, b, a, n_nodes);
}